// CustomModule_8065948582484
// MI455X (gfx1250) — compile-verified
//
#include <hip/hip_runtime.h>

// Problem constants (reference: score [16384, 16, 196] f32, n = 14)
#define F_FRAMES 16
#define P_POS    196
#define N_GRID   14
#define OUT_STRIDE (1 + F_FRAMES * P_POS)   // 3137 floats per batch

// Bit test on the 7-word (196-bit) mask at a *runtime* (wave-uniform) index.
// 7-way select tree keeps m[] in VGPRs (no scratch / dynamic indexing).
__device__ __forceinline__ unsigned mask_bit(const unsigned m[7], int p) {
    const int w = p >> 5;
    const unsigned sh = (unsigned)(p & 31);
    unsigned r = 0u;
#pragma unroll
    for (int k = 0; k < 7; ++k)
        r |= (k == w) ? ((m[k] >> sh) & 1u) : 0u;
    return r;
}

__device__ __forceinline__ void mask_set(unsigned m[7], int p) {
    const int w = p >> 5;
    const unsigned bit = 1u << (p & 31);
#pragma unroll
    for (int k = 0; k < 7; ++k)
        if (k == w) m[k] |= bit;
}

// One wave32 per batch: argmax over 196 via lane-strided loads + shfl_xor
// butterfly, then a uniform bit update of the 196-bit mask, then a coalesced
// non-temporal streaming write of the 196-float mask for this frame.
__global__ __launch_bounds__(256) void argmax_mask_scan(
    const float* __restrict__ score, float* __restrict__ out, int Bn)
{
    const int lane = threadIdx.x & 31;
    const int wave = threadIdx.x >> 5;
    const int b = blockIdx.x * (blockDim.x >> 5) + wave;
    if (b >= Bn) return;

    // Prior: m[r][c] = 1 for r in 4..13, c in 2..11 (constant-folds to 7 imms).
    unsigned m[7] = {0u, 0u, 0u, 0u, 0u, 0u, 0u};
#pragma unroll
    for (int r = 4; r < N_GRID; ++r)
#pragma unroll
        for (int c = 2; c < N_GRID - 2; ++c) {
            const int p = r * N_GRID + c;
            m[p >> 5] |= 1u << (p & 31);
        }

    const float* __restrict__ sb = score + (size_t)b * (F_FRAMES * P_POS);
    float* __restrict__ ob = out + (size_t)b * OUT_STRIDE;
    if (lane == 0) ob[0] = 1.0f;   // leading "ones" column

    for (int f = 0; f < F_FRAMES; ++f) {
        const float* __restrict__ sf = sb + f * P_POS;

        // Prefetch next frame's 784B while we reduce this one (gfx1250 path).
        if (f + 1 < F_FRAMES)
            __builtin_prefetch(sf + P_POS + lane * 8, 0, 1);

        // Lane-local argmax: p = lane, lane+32, ... (increasing p order keeps
        // first-occurrence tie semantics of jnp.argmax). lane < 32 <= 196, so
        // k = 0 is always valid.
        float bestv = __builtin_nontemporal_load(sf + lane);
        int   besti = lane;
#pragma unroll
        for (int k = 1; k < 7; ++k) {
            const int p = lane + k * 32;
            if (p < P_POS) {
                const float v = __builtin_nontemporal_load(sf + p);
                if (v > bestv) { bestv = v; besti = p; }
            }
        }

        // Wave32 butterfly reduction; ties -> lower index. Result is
        // broadcast to all lanes by construction.
#pragma unroll
        for (int off = 16; off >= 1; off >>= 1) {
            const float ov = __shfl_xor(bestv, off, 32);
            const int   oi = __shfl_xor(besti, off, 32);
            if (ov > bestv || (ov == bestv && oi < besti)) {
                bestv = ov; besti = oi;
            }
        }

        // Uniform mask update: set argmax cell iff a cross-neighbor is set.
        const int idx = besti;
        const int r = idx / N_GRID;
        const int c = idx - r * N_GRID;
        unsigned hit = 0u;
        if (r > 0)          hit |= mask_bit(m, idx - N_GRID);
        if (r < N_GRID - 1) hit |= mask_bit(m, idx + N_GRID);
        if (c > 0)          hit |= mask_bit(m, idx - 1);
        if (c < N_GRID - 1) hit |= mask_bit(m, idx + 1);
        if (hit) mask_set(m, idx);

        // Stream out this frame's mask as f32 0/1. For p = lane + k*32:
        // p>>5 == k and p&31 == lane, so indexing is compile-time constant.
        float* __restrict__ of = ob + 1 + f * P_POS;
#pragma unroll
        for (int k = 0; k < 7; ++k) {
            const int p = lane + k * 32;
            if (p < P_POS) {
                const float v = (float)((m[k] >> lane) & 1u);
                __builtin_nontemporal_store(v, of + p);
            }
        }
    }
}

extern "C" void kernel_launch(void* const* d_in, const int* in_sizes, int n_in,
                              void* d_out, int out_size, void* d_ws, size_t ws_size,
                              hipStream_t stream) {
    const float* score = (const float*)d_in[0];
    float* out = (float*)d_out;
    (void)d_ws; (void)ws_size; (void)out_size; (void)n_in;

    const int Bn = in_sizes[0] / (F_FRAMES * P_POS);   // 16384
    const int wavesPerBlock = 8;                        // 256 threads = 8 wave32
    dim3 block(32 * wavesPerBlock);
    dim3 grid((Bn + wavesPerBlock - 1) / wavesPerBlock);
    argmax_mask_scan<<<grid, block, 0, stream>>>(score, out, Bn);
}